// SlowROIPool_76785425318250
// MI455X (gfx1250) — compile-verified
//
#include <hip/hip_runtime.h>

// RoIPool (max) for MI455X / gfx1250.
// Bandwidth-bound gather -> use the Tensor Data Mover (TDM) to DMA each ROI's
// 3-D tile (w x h x channels) from global into LDS, then tiny max-reductions
// out of LDS with non-temporal output stores. WMMA is inapplicable (no GEMM
// structure in max pooling); TDM is the CDNA5-idiomatic data path here.

#define OUT_S   7
#define SCALE_F 0.0625f
#define B_      8
#define C_      256
#define H_      64
#define W_      64
#define N_      256
#define CB      64      // channels per block
#define CW      8       // channels per wave (8 waves * 8 = 64)
#define MAXHW   14      // max ROI extent in feature cells (ceil(200/16)+1)

typedef unsigned int v4u __attribute__((ext_vector_type(4)));
typedef int          v8i __attribute__((ext_vector_type(8)));
typedef int          v4i __attribute__((ext_vector_type(4)));

__device__ __forceinline__ int imin(int a, int b) { return a < b ? a : b; }
__device__ __forceinline__ int imax(int a, int b) { return a > b ? a : b; }

__global__ __launch_bounds__(256)
void roipool_tdm_kernel(const float* __restrict__ images,
                        const float* __restrict__ rois,
                        float* __restrict__ out)
{
    __shared__ float tile[CB * MAXHW * MAXHW];   // 50176 bytes

    const int n  = blockIdx.x;          // ROI index
    const int cb = blockIdx.y * CB;     // channel block base

    // ---- decode ROI, scalarize so descriptor dwords become SGPRs ----------
    const float* r = rois + (size_t)n * 5;
    int b  = __builtin_amdgcn_readfirstlane((int)r[0]);
    int x1 = __builtin_amdgcn_readfirstlane((int)floorf(r[1] * SCALE_F));
    int y1 = __builtin_amdgcn_readfirstlane((int)floorf(r[2] * SCALE_F));
    int x2 = __builtin_amdgcn_readfirstlane((int)ceilf (r[3] * SCALE_F));
    int y2 = __builtin_amdgcn_readfirstlane((int)ceilf (r[4] * SCALE_F));

    x1 = imin(imax(x1, 0), W_ - 1);
    y1 = imin(imax(y1, 0), H_ - 1);
    x2 = imin(imax(x2, x1 + 1), W_);
    y2 = imin(imax(y2, y1 + 1), H_);
    const int w = imin(x2 - x1, MAXHW);
    const int h = imin(y2 - y1, MAXHW);
    const int plane = w * h;

    // ---- per-wave TDM descriptor: 3-D tile  w x h x CW channels -----------
    const int wid = __builtin_amdgcn_readfirstlane((int)(threadIdx.x >> 5));
    const int cw  = cb + wid * CW;

    const unsigned lds_base = (unsigned)(uintptr_t)(void*)&tile[0];
    const unsigned lds_addr = lds_base + (unsigned)(wid * CW * plane * 4);

    const unsigned long long ga = (unsigned long long)(uintptr_t)
        (images + (((size_t)b * C_ + cw) * H_ + y1) * W_ + x1);

    // D# group0: count=1 | lds_addr | global_addr[56:0] | type=2
    v4u g0;
    g0.x = 1u;
    g0.y = lds_addr;
    g0.z = (unsigned)(ga & 0xffffffffu);
    g0.w = (unsigned)((ga >> 32) & 0x1ffffffu) | (2u << 30);

    // D# group1: data_size=4B; tensor 64x64; tile w x h x CW; strides 64, 4096
    v8i g1;
    g1[0] = (int)(2u << 16);                                  // data_size=2 (4B)
    g1[1] = (int)((unsigned)W_ << 16);                        // tensor_dim0 lo16
    g1[2] = (int)((unsigned)H_ << 16);                        // dim0 hi=0 | tensor_dim1 lo16
    g1[3] = (int)((unsigned)w  << 16);                        // dim1 hi=0 | tile_dim0
    g1[4] = (int)((unsigned)h | ((unsigned)CW << 16));        // tile_dim1 | tile_dim2
    g1[5] = W_;                                               // tensor_dim0_stride lo32
    g1[6] = (int)(((unsigned)(H_ * W_) & 0xffffu) << 16);     // s0 hi=0 | s1 lo16
    g1[7] = (int)((unsigned)(H_ * W_) >> 16);                 // tensor_dim1_stride hi32

    // D# group2: tensor_dim2 = C (z extent); no dim3, no iterate
    v4i g2; g2[0] = C_; g2[1] = 0; g2[2] = 0; g2[3] = 0;
    v4i g3; g3[0] = 0;  g3[1] = 0; g3[2] = 0; g3[3] = 0;
    // 6-arg toolchain form carries an extra (zero-filled) int32x8 operand.
    v8i g4; g4[0] = 0; g4[1] = 0; g4[2] = 0; g4[3] = 0;
            g4[4] = 0; g4[5] = 0; g4[6] = 0; g4[7] = 0;

    __builtin_amdgcn_tensor_load_to_lds(g0, g1, g2, g3, g4, 0);
    __builtin_amdgcn_s_wait_tensorcnt(0);
    __syncthreads();

    // ---- adaptive-max bins out of LDS -------------------------------------
    float* outn = out + ((size_t)n * C_ + cb) * (OUT_S * OUT_S);
    for (int idx = (int)threadIdx.x; idx < CB * OUT_S * OUT_S; idx += 256) {
        const int c   = idx / 49;
        const int bin = idx - c * 49;
        const int ph  = bin / 7;
        const int pw  = bin - ph * 7;
        const int lh = (ph * h) / 7, hh = ((ph + 1) * h + 6) / 7;
        const int lw = (pw * w) / 7, hw = ((pw + 1) * w + 6) / 7;
        const float* p = tile + c * plane;
        float m = -3.402823466e38f;
        for (int dy = lh; dy < hh; ++dy) {
            const float* row = p + dy * w;
            for (int dx = lw; dx < hw; ++dx)
                m = fmaxf(m, row[dx]);
        }
        __builtin_nontemporal_store(m, &outn[(size_t)c * 49 + bin]);
    }
}

extern "C" void kernel_launch(void* const* d_in, const int* in_sizes, int n_in,
                              void* d_out, int out_size, void* d_ws, size_t ws_size,
                              hipStream_t stream) {
    const float* images = (const float*)d_in[0];   // [8,256,64,64] f32
    const float* rois   = (const float*)d_in[1];   // [256,5] f32
    float* out = (float*)d_out;                    // [256,256,7,7] f32
    (void)in_sizes; (void)n_in; (void)out_size; (void)d_ws; (void)ws_size;

    dim3 grid(N_, C_ / CB);
    dim3 block(256);
    roipool_tdm_kernel<<<grid, block, 0, stream>>>(images, rois, out);
}